// MemPool_81973745811865
// MI455X (gfx1250) — compile-verified
//
#include <hip/hip_runtime.h>
#include <hip/hip_bf16.h>
#include <math.h>

#define NTOT   65536
#define BGR    64
#define NMAXN  1024
#define EEDGE  1048576
#define ETOT   (EEDGE + NTOT)
#define IND    128
#define HIDD   80
#define HEADSN 5
#define K1N    10
#define OUTN   10

typedef float v2f __attribute__((ext_vector_type(2)));
typedef float v8f __attribute__((ext_vector_type(8)));

// ---------------------------------------------------------------------------
// Generic fp32 WMMA GEMM:  C[M,N] = act(A[M,K] @ B(+bias))
//   - V_WMMA_F32_16X16X4_F32, one 16x16 output tile per wave, K stepped by 4
//   - TRANSB: B stored [N,K] row-major (used for xd @ keys^T)
//   - N may be non-multiple of 16: out-of-range lanes clamp their B column to
//     N-1 (their output columns are never stored), keeping all loads in-bounds
//     and EXEC full (no divergence) through the WMMA loop.
//   - M must be a multiple of 16, K a multiple of 4
// ---------------------------------------------------------------------------
template <bool TRANSB, bool BIAS, int ACT>
__global__ void gemm_wmma_kernel(const float* __restrict__ A,
                                 const float* __restrict__ B,
                                 float* __restrict__ C,
                                 const float* __restrict__ bias,
                                 int M, int N, int K, int ldC)
{
  const int lane    = threadIdx.x & 31;
  const int waveId  = threadIdx.x >> 5;
  const int rowTile = blockIdx.x * (blockDim.x >> 5) + waveId;
  const int MT      = M >> 4;
  if (rowTile >= MT) return;
  const int rowBase = rowTile << 4;
  const int colBase = blockIdx.y << 4;

  const bool hi   = lane >= 16;
  const int  r    = lane & 15;
  const int  col  = colBase + r;
  const int  colc = col < N ? col : (N - 1);   // clamp: keeps loads in-bounds

  v8f c;
#pragma unroll
  for (int v = 0; v < 8; ++v) c[v] = 0.0f;

  const float* aPtr = A + (size_t)(rowBase + r) * (size_t)K + (hi ? 2 : 0);
  const float* bPtr;
  if (TRANSB) bPtr = B + (size_t)colc * (size_t)K + (hi ? 2 : 0);
  else        bPtr = B + (size_t)(hi ? 2 : 0) * (size_t)N + colc;

  for (int k = 0; k < K; k += 4) {
    const v2f a = *(const v2f*)(aPtr + k);     // global_load_b64, 8B aligned
    v2f b;
    if (TRANSB) {
      b = *(const v2f*)(bPtr + k);             // global_load_b64, 8B aligned
    } else {
      b.x = bPtr[(size_t)k * N];
      b.y = bPtr[(size_t)(k + 1) * N];
    }
    // (neg_a, A, neg_b, B, c_mod, C, reuse_a, reuse_b)
    c = __builtin_amdgcn_wmma_f32_16x16x4_f32(false, a, false, b, (short)0, c,
                                              false, false);
  }

  if (col < N) {
    const float bv = BIAS ? bias[col] : 0.0f;
#pragma unroll
    for (int v = 0; v < 8; ++v) {
      const int row = rowBase + v + (hi ? 8 : 0);
      float val = c[v] + bv;
      if (ACT == 1) val = val > 0.0f ? val : 0.01f * val;
      C[(size_t)row * (size_t)ldC + col] = val;
    }
  }
}

// ---------------------------------------------------------------------------
// Per-node attention dots: hs[i] = h[i,:]·a_src, hd[i] = h[i,:]·a_dst
// one wave per node (80 = 32 + 32 + 16 lanes)
// ---------------------------------------------------------------------------
__global__ void node_dots_kernel(const float* __restrict__ h,
                                 const float* __restrict__ asrc,
                                 const float* __restrict__ adst,
                                 float* __restrict__ hs,
                                 float* __restrict__ hd)
{
  const int lane = threadIdx.x & 31;
  const int node = blockIdx.x * (blockDim.x >> 5) + (threadIdx.x >> 5);
  if (node >= NTOT) return;
  const size_t base = (size_t)node * HIDD;
  float s1 = 0.f, s2 = 0.f;
  float v0 = h[base + lane];       s1 += v0 * asrc[lane];      s2 += v0 * adst[lane];
  float v1 = h[base + 32 + lane];  s1 += v1 * asrc[32 + lane]; s2 += v1 * adst[32 + lane];
  if (lane < 16) {
    float v2 = h[base + 64 + lane]; s1 += v2 * asrc[64 + lane]; s2 += v2 * adst[64 + lane];
  }
  for (int off = 16; off >= 1; off >>= 1) {
    s1 += __shfl_xor(s1, off, 32);
    s2 += __shfl_xor(s2, off, 32);
  }
  if (lane == 0) { hs[node] = s1; hd[node] = s2; }
}

__global__ void fill_kernel(float* __restrict__ p, float v, int n)
{
  const int i = blockIdx.x * blockDim.x + threadIdx.x;
  if (i < n) p[i] = v;
}

// e = leaky0.2(hs[src]+hd[dst]); atomic segment-max into mmax[dst]
__global__ void edge_logits_kernel(const int* __restrict__ ei,
                                   const float* __restrict__ hs,
                                   const float* __restrict__ hd,
                                   float* __restrict__ e,
                                   float* __restrict__ mmax)
{
  const int i = blockIdx.x * blockDim.x + threadIdx.x;
  if (i >= ETOT) return;
  int s, d;
  if (i < EEDGE) { s = ei[i]; d = ei[EEDGE + i]; } else { s = d = i - EEDGE; }
  float v = hs[s] + hd[d];
  v = v > 0.0f ? v : 0.2f * v;
  e[i] = v;
  int* ia = (int*)(mmax + d);
  int old = __float_as_int(mmax[d]);
  while (__int_as_float(old) < v) {
    const int assumed = old;
    old = atomicCAS(ia, assumed, __float_as_int(v));
    if (old == assumed) break;
  }
}

__global__ void edge_weight_kernel(const int* __restrict__ ei,
                                   float* __restrict__ e,
                                   const float* __restrict__ mmax,
                                   float* __restrict__ denom)
{
  const int i = blockIdx.x * blockDim.x + threadIdx.x;
  if (i >= ETOT) return;
  const int d = (i < EEDGE) ? ei[EEDGE + i] : (i - EEDGE);
  const float w = expf(e[i] - mmax[d]);
  e[i] = w;
  atomicAdd(denom + d, w);
}

__global__ void edge_norm_kernel(const int* __restrict__ ei,
                                 float* __restrict__ e,
                                 const float* __restrict__ denom)
{
  const int i = blockIdx.x * blockDim.x + threadIdx.x;
  if (i >= ETOT) return;
  const int d = (i < EEDGE) ? ei[EEDGE + i] : (i - EEDGE);
  e[i] = e[i] / denom[d];
}

// out[dst,f] += h[src,f] * alpha  (one thread per (edge, feature))
__global__ void agg_kernel(const int* __restrict__ ei,
                           const float* __restrict__ h,
                           const float* __restrict__ alpha,
                           float* __restrict__ out)
{
  const long long idx = (long long)blockIdx.x * blockDim.x + threadIdx.x;
  if (idx >= (long long)ETOT * HIDD) return;
  const int i = (int)(idx / HIDD);
  const int f = (int)(idx % HIDD);
  int s, d;
  if (i < EEDGE) { s = ei[i]; d = ei[EEDGE + i]; } else { s = d = i - EEDGE; }
  atomicAdd(out + (size_t)d * HIDD + f, h[(size_t)s * HIDD + f] * alpha[i]);
}

__global__ void bias_act_kernel(float* __restrict__ p,
                                const float* __restrict__ b, int n)
{
  const int i = blockIdx.x * blockDim.x + threadIdx.x;
  if (i >= n) return;
  const float v = p[i] + b[i % HIDD];
  p[i] = v > 0.0f ? v : 0.01f * v;
}

__global__ void knorm_kernel(const float* __restrict__ keys,
                             float* __restrict__ knorm, int rows)
{
  const int j = blockIdx.x * blockDim.x + threadIdx.x;
  if (j >= rows) return;
  float s = 0.f;
  for (int f = 0; f < HIDD; ++f) { const float v = keys[j * HIDD + f]; s += v * v; }
  knorm[j] = s;
}

// MemPool1 assignment: d -> t-student -> head mix -> softmax  => S [NTOT, 10]
__global__ void assign_kernel(const float* __restrict__ xd,
                              const float* __restrict__ G,
                              const float* __restrict__ knorm,
                              const float* __restrict__ convw,
                              float* __restrict__ S)
{
  const int node = blockIdx.x * blockDim.x + threadIdx.x;
  if (node >= NTOT) return;
  const float* xr = xd + (size_t)node * HIDD;
  float xn = 0.f;
  for (int f = 0; f < HIDD; ++f) xn += xr[f] * xr[f];
  const float* g = G + (size_t)node * (HEADSN * K1N);

  float t[K1N];
#pragma unroll
  for (int k = 0; k < K1N; ++k) t[k] = 0.f;
  for (int hh = 0; hh < HEADSN; ++hh) {
    float inv[K1N]; float rs = 0.f;
#pragma unroll
    for (int k = 0; k < K1N; ++k) {
      float dd = xn + knorm[hh * K1N + k] - 2.0f * g[hh * K1N + k];
      dd = dd > 0.f ? dd : 0.f;
      const float iv = 1.0f / (1.0f + dd);
      inv[k] = iv; rs += iv;
    }
    const float cw = convw[hh] / rs;
#pragma unroll
    for (int k = 0; k < K1N; ++k) t[k] += cw * inv[k];
  }
  float mx = t[0];
#pragma unroll
  for (int k = 1; k < K1N; ++k) mx = fmaxf(mx, t[k]);
  float se = 0.f;
#pragma unroll
  for (int k = 0; k < K1N; ++k) { const float ev = expf(t[k] - mx); t[k] = ev; se += ev; }
  const float rinv = 1.0f / se;
#pragma unroll
  for (int k = 0; k < K1N; ++k) S[(size_t)node * K1N + k] = t[k] * rinv;
}

// xp[b,k,f] = sum_n S[b,n,k] * xd[b,n,f]   (one block per graph)
__global__ void xp_kernel(const float* __restrict__ S,
                          const float* __restrict__ xd,
                          float* __restrict__ xp)
{
  const int b = blockIdx.x;
  for (int o = threadIdx.x; o < K1N * HIDD; o += blockDim.x) {
    const int k = o / HIDD, f = o % HIDD;
    const float* Sb = S  + (size_t)b * NMAXN * K1N  + k;
    const float* Xb = xd + (size_t)b * NMAXN * HIDD + f;
    float acc = 0.f;
    for (int n = 0; n < NMAXN; ++n)
      acc += Sb[(size_t)n * K1N] * Xb[(size_t)n * HIDD];
    xp[((size_t)b * K1N + k) * HIDD + f] = acc;
  }
}

__global__ void colsum_kernel(const float* __restrict__ S, float* __restrict__ cs)
{
  const int j = blockIdx.x * blockDim.x + threadIdx.x;
  if (j >= BGR * K1N) return;
  const int b = j / K1N, k = j % K1N;
  const float* Sb = S + (size_t)b * NMAXN * K1N + k;
  float s = 0.f;
  for (int n = 0; n < NMAXN; ++n) s += Sb[(size_t)n * K1N];
  cs[j] = s;
}

__global__ void kl_kernel(const float* __restrict__ S,
                          const float* __restrict__ cs,
                          float* __restrict__ klacc)
{
  const int node = blockIdx.x * blockDim.x + threadIdx.x;
  if (node >= NTOT) return;
  const int b = node >> 10;
  const float* Sr = S + (size_t)node * K1N;
  float P[K1N]; float rs = 0.f;
#pragma unroll
  for (int k = 0; k < K1N; ++k) {
    const float s = Sr[k];
    const float p = s * s / cs[b * K1N + k];
    P[k] = p; rs += p;
  }
  const float rrs = 1.0f / rs;
  float acc = 0.f;
#pragma unroll
  for (int k = 0; k < K1N; ++k) {
    float p = fmaxf(P[k] * rrs, 1e-15f);
    const float s = fmaxf(Sr[k], 1e-15f);
    acc += p * (logf(p) - logf(s));
  }
  atomicAdd(klacc, acc);
}

// MemPool2 (S2==1 exactly, kl2==0 exactly) + MLP head + BatchNorm, one block
__global__ void head_kernel(const float* __restrict__ x1,
                            const float* __restrict__ linw2,
                            const float* __restrict__ linb2,
                            const float* __restrict__ mlpw1,
                            const float* __restrict__ mlpb1,
                            const float* __restrict__ gamma,
                            const float* __restrict__ beta,
                            const float* __restrict__ mlpw2,
                            const float* __restrict__ mlpb2,
                            const float* __restrict__ klacc,
                            float* __restrict__ out)
{
  __shared__ float sA[BGR * HIDD];
  __shared__ float sB[BGR * HIDD];
  __shared__ float mu[HIDD], rsig[HIDD];
  const int t = threadIdx.x;

  // xp2[b,f] = sum over 10 clusters of x1
  for (int idx = t; idx < BGR * HIDD; idx += blockDim.x) {
    const int b = idx / HIDD, f = idx % HIDD;
    float s = 0.f;
    for (int n = 0; n < K1N; ++n) s += x1[((size_t)b * K1N + n) * HIDD + f];
    sA[idx] = s;
  }
  __syncthreads();
  // x2 = leaky(xp2 @ linw2 + linb2)
  for (int idx = t; idx < BGR * HIDD; idx += blockDim.x) {
    const int b = idx / HIDD, f = idx % HIDD;
    float s = linb2[f];
    for (int k = 0; k < HIDD; ++k) s += sA[b * HIDD + k] * linw2[k * HIDD + f];
    sB[idx] = s > 0.f ? s : 0.01f * s;
  }
  __syncthreads();
  // z = x2 @ mlp_w1 + mlp_b1
  for (int idx = t; idx < BGR * HIDD; idx += blockDim.x) {
    const int b = idx / HIDD, f = idx % HIDD;
    float s = mlpb1[f];
    for (int k = 0; k < HIDD; ++k) s += sB[b * HIDD + k] * mlpw1[k * HIDD + f];
    sA[idx] = s;
  }
  __syncthreads();
  // BatchNorm stats (biased var, training mode)
  if (t < HIDD) {
    float m = 0.f;
    for (int b = 0; b < BGR; ++b) m += sA[b * HIDD + t];
    m *= (1.0f / BGR);
    float v = 0.f;
    for (int b = 0; b < BGR; ++b) { const float dd = sA[b * HIDD + t] - m; v += dd * dd; }
    v *= (1.0f / BGR);
    mu[t] = m; rsig[t] = rsqrtf(v + 1e-5f);
  }
  __syncthreads();
  for (int idx = t; idx < BGR * HIDD; idx += blockDim.x) {
    const int f = idx % HIDD;
    const float zn = (sA[idx] - mu[f]) * rsig[f] * gamma[f] + beta[f];
    sB[idx] = zn > 0.f ? zn : 0.f;
  }
  __syncthreads();
  for (int idx = t; idx < BGR * OUTN; idx += blockDim.x) {
    const int b = idx / OUTN, o = idx % OUTN;
    float s = mlpb2[o];
    for (int k = 0; k < HIDD; ++k) s += sB[b * HIDD + k] * mlpw2[k * OUTN + o];
    out[idx] = s;
  }
  if (t == 0) out[BGR * OUTN] = klacc[0] * (1.0f / BGR);  // kl2 == 0 exactly
}

// ---------------------------------------------------------------------------
extern "C" void kernel_launch(void* const* d_in, const int* in_sizes, int n_in,
                              void* d_out, int out_size, void* d_ws, size_t ws_size,
                              hipStream_t stream)
{
  const float* x      = (const float*)d_in[0];
  const int*   ei     = (const int*)d_in[1];
  // d_in[2] (batch) unused: equal contiguous partition -> pure reshape
  const float* W1     = (const float*)d_in[3];
  const float* a_src1 = (const float*)d_in[4];
  const float* a_dst1 = (const float*)d_in[5];
  const float* b1     = (const float*)d_in[6];
  const float* W2     = (const float*)d_in[7];
  const float* a_src2 = (const float*)d_in[8];
  const float* a_dst2 = (const float*)d_in[9];
  const float* b2     = (const float*)d_in[10];
  const float* keys1  = (const float*)d_in[11];
  const float* convw1 = (const float*)d_in[12];
  const float* linw1  = (const float*)d_in[13];
  const float* linb1  = (const float*)d_in[14];
  // keys2(15)/convw2(16) mathematically dead (K2=1 -> S2==1)
  const float* linw2  = (const float*)d_in[17];
  const float* linb2  = (const float*)d_in[18];
  const float* mlpw1  = (const float*)d_in[19];
  const float* mlpb1  = (const float*)d_in[20];
  const float* gamma  = (const float*)d_in[21];
  const float* beta   = (const float*)d_in[22];
  const float* mlpw2  = (const float*)d_in[23];
  const float* mlpb2  = (const float*)d_in[24];
  float* out = (float*)d_out;

  float* ws = (float*)d_ws;
  size_t off = 0;
  float* bufA  = ws + off; off += (size_t)NTOT * HIDD;   // h1 / h2 / G,S scratch
  float* bufB  = ws + off; off += (size_t)NTOT * HIDD;   // agg1/act -> agg2/xd
  float* hs    = ws + off; off += NTOT;
  float* hd    = ws + off; off += NTOT;
  float* mmax  = ws + off; off += NTOT;
  float* denom = ws + off; off += NTOT;
  float* alpha = ws + off; off += ETOT;
  float* knrm  = ws + off; off += 64;
  float* cs    = ws + off; off += 640;
  float* klacc = ws + off; off += 16;
  float* xp    = ws + off; off += (size_t)BGR * K1N * HIDD;
  float* x1    = ws + off; off += (size_t)BGR * K1N * HIDD;
  if (ws_size < off * sizeof(float)) return;

  float* G = bufA;                           // [NTOT, 50]
  float* S = bufA + (size_t)NTOT * 50;       // [NTOT, 10]

  const dim3 blk(256);
  auto cdiv = [](long long a, long long b) { return (unsigned)((a + b - 1) / b); };
  const unsigned egrid  = cdiv(ETOT, 256);
  const unsigned efgrid = cdiv((long long)ETOT * HIDD, 256);
  const unsigned nfgrid = cdiv((long long)NTOT * HIDD, 256);

  // ---- GAT layer 1 ----
  gemm_wmma_kernel<false, false, 0><<<dim3(NTOT / 16 / 8, 5), blk, 0, stream>>>(
      x, W1, bufA, nullptr, NTOT, HIDD, IND, HIDD);
  node_dots_kernel<<<dim3(NTOT / 8), blk, 0, stream>>>(bufA, a_src1, a_dst1, hs, hd);
  fill_kernel<<<dim3(cdiv(NTOT, 256)), blk, 0, stream>>>(mmax, -3.0e38f, NTOT);
  fill_kernel<<<dim3(cdiv(NTOT, 256)), blk, 0, stream>>>(denom, 0.0f, NTOT);
  fill_kernel<<<dim3(nfgrid), blk, 0, stream>>>(bufB, 0.0f, NTOT * HIDD);
  edge_logits_kernel<<<dim3(egrid), blk, 0, stream>>>(ei, hs, hd, alpha, mmax);
  edge_weight_kernel<<<dim3(egrid), blk, 0, stream>>>(ei, alpha, mmax, denom);
  edge_norm_kernel<<<dim3(egrid), blk, 0, stream>>>(ei, alpha, denom);
  agg_kernel<<<dim3(efgrid), blk, 0, stream>>>(ei, bufA, alpha, bufB);
  bias_act_kernel<<<dim3(nfgrid), blk, 0, stream>>>(bufB, b1, NTOT * HIDD);

  // ---- GAT layer 2 ----
  gemm_wmma_kernel<false, false, 0><<<dim3(NTOT / 16 / 8, 5), blk, 0, stream>>>(
      bufB, W2, bufA, nullptr, NTOT, HIDD, HIDD, HIDD);
  node_dots_kernel<<<dim3(NTOT / 8), blk, 0, stream>>>(bufA, a_src2, a_dst2, hs, hd);
  fill_kernel<<<dim3(cdiv(NTOT, 256)), blk, 0, stream>>>(mmax, -3.0e38f, NTOT);
  fill_kernel<<<dim3(cdiv(NTOT, 256)), blk, 0, stream>>>(denom, 0.0f, NTOT);
  fill_kernel<<<dim3(nfgrid), blk, 0, stream>>>(bufB, 0.0f, NTOT * HIDD);
  edge_logits_kernel<<<dim3(egrid), blk, 0, stream>>>(ei, hs, hd, alpha, mmax);
  edge_weight_kernel<<<dim3(egrid), blk, 0, stream>>>(ei, alpha, mmax, denom);
  edge_norm_kernel<<<dim3(egrid), blk, 0, stream>>>(ei, alpha, denom);
  agg_kernel<<<dim3(efgrid), blk, 0, stream>>>(ei, bufA, alpha, bufB);
  bias_act_kernel<<<dim3(nfgrid), blk, 0, stream>>>(bufB, b2, NTOT * HIDD);
  // bufB == xd [64,1024,80] (dense batch is a pure reshape, mask all ones)

  // ---- MemPooling 1 ----
  gemm_wmma_kernel<true, false, 0><<<dim3(NTOT / 16 / 8, 4), blk, 0, stream>>>(
      bufB, keys1, G, nullptr, NTOT, HEADSN * K1N, HIDD, HEADSN * K1N);
  knorm_kernel<<<dim3(1), dim3(64), 0, stream>>>(keys1, knrm, HEADSN * K1N);
  assign_kernel<<<dim3(NTOT / 256), blk, 0, stream>>>(bufB, G, knrm, convw1, S);
  colsum_kernel<<<dim3(cdiv(BGR * K1N, 256)), blk, 0, stream>>>(S, cs);
  fill_kernel<<<dim3(1), dim3(16), 0, stream>>>(klacc, 0.0f, 16);
  kl_kernel<<<dim3(NTOT / 256), blk, 0, stream>>>(S, cs, klacc);
  xp_kernel<<<dim3(BGR), blk, 0, stream>>>(S, bufB, xp);
  gemm_wmma_kernel<false, true, 1><<<dim3(5, 5), blk, 0, stream>>>(
      xp, linw1, x1, linb1, BGR * K1N, HIDD, HIDD, HIDD);

  // ---- MemPooling 2 (degenerate) + MLP head + BN ----
  head_kernel<<<dim3(1), blk, 0, stream>>>(x1, linw2, linb2, mlpw1, mlpb1,
                                           gamma, beta, mlpw2, mlpb2, klacc, out);
}